// TSA_Transform_6201932775522
// MI455X (gfx1250) — compile-verified
//
#include <hip/hip_runtime.h>

// ---------------------------------------------------------------------------
// TSA transform forward for MI455X (gfx1250, wave32).
// WMMA fp32 16x16x4 with 2-way N register blocking (16x32 tile per wave):
// A fragment loaded once per k-step, two WMMAs issued -> halves A traffic.
// ---------------------------------------------------------------------------

#define B_    2
#define H_    256
#define W_    256
#define CIN_  28
#define C_    28
#define NH_   8
#define UA_   96
#define NB_   16          // NH_*B_
#define C2_   56          // 2*C_
#define HW_   65536       // H_*W_
#define EPS_  1e-5f
#define PI_   3.14159265358979323846f

typedef float v2f __attribute__((ext_vector_type(2)));
typedef float v8f __attribute__((ext_vector_type(8)));

__device__ __forceinline__ float bn_apply(float x, const float* sc, const float* bi,
                                          const float* mn, const float* vr, int c) {
  return (x - mn[c]) * (sc[c] * rsqrtf(vr[c] + EPS_)) + bi[c];
}

__device__ __forceinline__ v8f wmma4(v2f a, v2f b, v8f c) {
  // D = A(16x4 f32) * B(4x16 f32) + C(16x16 f32)
  return __builtin_amdgcn_wmma_f32_16x16x4_f32(false, a, false, b, (short)0, c,
                                               false, false);
}

// ===========================================================================
// 1) preproc: 1x1 conv CIN->C + BN          feat[B,C,H,W]
// ===========================================================================
__global__ void k_preproc(const float* __restrict__ img, const float* __restrict__ w,
                          const float* sc, const float* bi, const float* mn, const float* vr,
                          float* __restrict__ feat) {
  int idx = blockIdx.x * blockDim.x + threadIdx.x;
  if (idx >= B_ * C_ * HW_) return;
  int p = idx % HW_;
  int c = (idx / HW_) % C_;
  int b = idx / (HW_ * C_);
  float acc = 0.f;
  #pragma unroll 4
  for (int ci = 0; ci < CIN_; ++ci)
    acc += img[(b * CIN_ + ci) * HW_ + p] * w[c * CIN_ + ci];
  feat[idx] = bn_apply(acc, sc, bi, mn, vr, c);
}

// ===========================================================================
// 2) spatial branch: conv1 (1x5, s=(1,2), p=(0,2), C->NH) + BN + relu
// ===========================================================================
__global__ void k_sp_conv1(const float* __restrict__ feat, const float* __restrict__ w,
                           const float* cb, const float* sc, const float* bi,
                           const float* mn, const float* vr,
                           float* __restrict__ outp, int transpose) {
  int idx = blockIdx.x * blockDim.x + threadIdx.x;
  if (idx >= B_ * NH_ * 256 * 128) return;
  int j = idx % 128;
  int i = (idx / 128) % 256;
  int o = (idx / (128 * 256)) % NH_;
  int b = idx / (128 * 256 * NH_);
  float acc = cb[o];
  for (int ci = 0; ci < C_; ++ci) {
    const float* fp = feat + (b * C_ + ci) * HW_;
    const float* wp = w + (o * C_ + ci) * 5;
    #pragma unroll
    for (int kw = 0; kw < 5; ++kw) {
      int wi = 2 * j + kw - 2;
      if (wi < 0 || wi >= 256) continue;
      float xv = transpose ? fp[wi * 256 + i] : fp[i * 256 + wi];
      acc += xv * wp[kw];
    }
  }
  outp[idx] = fmaxf(bn_apply(acc, sc, bi, mn, vr, o), 0.f);
}

// conv2 (3x5, s=(2,2), p=(1,2), NH->NH) + BN + relu: out [B,NH,128,64]
__global__ void k_sp_conv2(const float* __restrict__ inp, const float* __restrict__ w,
                           const float* cb, const float* sc, const float* bi,
                           const float* mn, const float* vr, float* __restrict__ outp) {
  int idx = blockIdx.x * blockDim.x + threadIdx.x;
  if (idx >= B_ * NH_ * 128 * 64) return;
  int j = idx % 64;
  int i = (idx / 64) % 128;
  int o = (idx / (64 * 128)) % NH_;
  int b = idx / (64 * 128 * NH_);
  float acc = cb[o];
  for (int ci = 0; ci < NH_; ++ci) {
    #pragma unroll
    for (int kh = 0; kh < 3; ++kh) {
      int hi = 2 * i + kh - 1;
      if (hi < 0 || hi >= 256) continue;
      #pragma unroll
      for (int kw = 0; kw < 5; ++kw) {
        int wi = 2 * j + kw - 2;
        if (wi < 0 || wi >= 128) continue;
        acc += inp[((b * NH_ + ci) * 256 + hi) * 128 + wi] * w[((o * NH_ + ci) * 3 + kh) * 5 + kw];
      }
    }
  }
  outp[idx] = fmaxf(bn_apply(acc, sc, bi, mn, vr, o), 0.f);
}

// spatial FC via WMMA: in [2048,64], fw [96,64]; out head-major [16,128,96]
// 2-way N blocking: 128 M-tiles * 3 N-pairs
__global__ void k_sp_fc_wmma(const float* __restrict__ inp, const float* __restrict__ fw,
                             const float* __restrict__ fb, float* __restrict__ outp) {
  int wid = blockIdx.x;
  int mt = wid / 3, np = wid % 3;
  int lane = threadIdx.x & 31;
  int half = lane >> 4, l16 = lane & 15;
  int arow = mt * 16 + l16;
  int bcol0 = np * 32 + l16;
  int bcol1 = bcol0 + 16;
  v8f acc0 = {0.f, 0.f, 0.f, 0.f, 0.f, 0.f, 0.f, 0.f};
  v8f acc1 = acc0;
  for (int k0 = 0; k0 < 64; k0 += 4) {
    int ka = k0 + 2 * half;
    v2f a, b0, b1;
    a.x = inp[arow * 64 + ka];
    a.y = inp[arow * 64 + ka + 1];
    b0.x = fw[bcol0 * 64 + ka];
    b0.y = fw[bcol0 * 64 + ka + 1];
    b1.x = fw[bcol1 * 64 + ka];
    b1.y = fw[bcol1 * 64 + ka + 1];
    acc0 = wmma4(a, b0, acc0);
    acc1 = wmma4(a, b1, acc1);
  }
  #pragma unroll
  for (int j = 0; j < 8; ++j) {
    int mm = mt * 16 + j + 8 * half;
    int b = mm >> 10, h = (mm >> 7) & 7, s = mm & 127;
    long base = ((h * B_ + b) * 128 + s) * (long)UA_;
    int nn0 = np * 32 + l16;
    outp[base + nn0] = fmaxf(acc0[j] + fb[nn0], 0.f);
    outp[base + nn0 + 16] = fmaxf(acc1[j] + fb[nn0 + 16], 0.f);
  }
}

// ===========================================================================
// 3) spectral convs (5x5, stride 2, valid) + BN + relu (generic over size)
// ===========================================================================
__global__ void k_spec_conv(const float* __restrict__ inp, int in_sz, int out_sz,
                            const float* __restrict__ w, const float* cb,
                            const float* sc, const float* bi, const float* mn,
                            const float* vr, float* __restrict__ outp) {
  int idx = blockIdx.x * blockDim.x + threadIdx.x;
  int total = B_ * C_ * out_sz * out_sz;
  if (idx >= total) return;
  int j = idx % out_sz;
  int i = (idx / out_sz) % out_sz;
  int o = (idx / (out_sz * out_sz)) % C_;
  int b = idx / (out_sz * out_sz * C_);
  float acc = cb[o];
  for (int ci = 0; ci < C_; ++ci) {
    const float* ip = inp + (b * C_ + ci) * in_sz * in_sz;
    const float* wp = w + (o * C_ + ci) * 25;
    #pragma unroll
    for (int kh = 0; kh < 5; ++kh)
      #pragma unroll
      for (int kw = 0; kw < 5; ++kw)
        acc += ip[(2 * i + kh) * in_sz + (2 * j + kw)] * wp[kh * 5 + kw];
  }
  outp[idx] = fmaxf(bn_apply(acc, sc, bi, mn, vr, o), 0.f);
}

// spectral FC via WMMA: A [56,3721], fw [768,3721]; out head-major [16,28,96]
// 4 M-tiles * 24 N-pairs
__global__ void k_spec_fc_wmma(const float* __restrict__ t2, const float* __restrict__ fw,
                               const float* __restrict__ fb, float* __restrict__ outp) {
  const int K = 3721;
  int wid = blockIdx.x;
  int mt = wid / 24, np = wid % 24;
  int lane = threadIdx.x & 31;
  int half = lane >> 4, l16 = lane & 15;
  int arow = mt * 16 + l16;
  bool arok = arow < 56;
  int bcol0 = np * 32 + l16;
  int bcol1 = bcol0 + 16;
  v8f acc0 = {0.f, 0.f, 0.f, 0.f, 0.f, 0.f, 0.f, 0.f};
  v8f acc1 = acc0;
  for (int k0 = 0; k0 < 3724; k0 += 4) {
    int ka = k0 + 2 * half;
    v2f a, b0, b1;
    a.x = (arok && ka < K) ? t2[arow * K + ka] : 0.f;
    a.y = (arok && ka + 1 < K) ? t2[arow * K + ka + 1] : 0.f;
    b0.x = (ka < K) ? fw[bcol0 * K + ka] : 0.f;
    b0.y = (ka + 1 < K) ? fw[bcol0 * K + ka + 1] : 0.f;
    b1.x = (ka < K) ? fw[bcol1 * K + ka] : 0.f;
    b1.y = (ka + 1 < K) ? fw[bcol1 * K + ka + 1] : 0.f;
    acc0 = wmma4(a, b0, acc0);
    acc1 = wmma4(a, b1, acc1);
  }
  #pragma unroll
  for (int j = 0; j < 8; ++j) {
    int mm = mt * 16 + j + 8 * half;
    if (mm >= 56) continue;
    int b = mm / C_, c = mm % C_;
    #pragma unroll
    for (int q = 0; q < 2; ++q) {
      int nn = np * 32 + l16 + q * 16;
      int h = nn / UA_, u = nn % UA_;
      float v = (q == 0 ? acc0[j] : acc1[j]) + fb[nn];
      outp[((h * B_ + b) * C_ + c) * UA_ + u] = fmaxf(v, 0.f);
    }
  }
}

// ===========================================================================
// 4) value: 1x1 conv C->NH*C + BN, written head-major v[16,28,256,256]
// ===========================================================================
__global__ void k_value(const float* __restrict__ feat, const float* __restrict__ w,
                        const float* sc, const float* bi, const float* mn,
                        const float* vr, float* __restrict__ vout) {
  int idx = blockIdx.x * blockDim.x + threadIdx.x;
  if (idx >= NB_ * C_ * HW_) return;
  int p = idx % HW_;
  int c = (idx / HW_) % C_;
  int n = idx / (HW_ * C_);
  int h = n / B_, b = n % B_;
  int ch = h * C_ + c;
  float acc = 0.f;
  #pragma unroll 4
  for (int ci = 0; ci < C_; ++ci)
    acc += feat[(b * C_ + ci) * HW_ + p] * w[ch * C_ + ci];
  vout[idx] = bn_apply(acc, sc, bi, mn, vr, ch);
}

// ===========================================================================
// 5) attention scores Q@K^T / sqrt(UA): [16,128,96] -> [16,128,128]
//    16 batches * (8 M-tiles * 4 N-pairs)
// ===========================================================================
__global__ void k_scores_wmma(const float* __restrict__ q, const float* __restrict__ kk,
                              float* __restrict__ attn) {
  int wid = blockIdx.x;
  int n = wid >> 5;
  int t = wid & 31;
  int mt = t >> 2, np = t & 3;
  int lane = threadIdx.x & 31;
  int half = lane >> 4, l16 = lane & 15;
  const float* qn = q + n * 128 * UA_;
  const float* kn = kk + n * 128 * UA_;
  int arow = mt * 16 + l16;
  int bcol0 = np * 32 + l16;
  int bcol1 = bcol0 + 16;
  v8f acc0 = {0.f, 0.f, 0.f, 0.f, 0.f, 0.f, 0.f, 0.f};
  v8f acc1 = acc0;
  for (int k0 = 0; k0 < UA_; k0 += 4) {
    int ka = k0 + 2 * half;
    v2f a, b0, b1;
    a.x = qn[arow * UA_ + ka];
    a.y = qn[arow * UA_ + ka + 1];
    b0.x = kn[bcol0 * UA_ + ka];
    b0.y = kn[bcol0 * UA_ + ka + 1];
    b1.x = kn[bcol1 * UA_ + ka];
    b1.y = kn[bcol1 * UA_ + ka + 1];
    acc0 = wmma4(a, b0, acc0);
    acc1 = wmma4(a, b1, acc1);
  }
  const float inv_scale = 0.10206207261596576f;   // 1/sqrt(96)
  #pragma unroll
  for (int j = 0; j < 8; ++j) {
    int mm = mt * 16 + j + 8 * half;
    int nn = np * 32 + l16;
    attn[n * 16384 + mm * 128 + nn] = acc0[j] * inv_scale;
    attn[n * 16384 + mm * 128 + nn + 16] = acc1[j] * inv_scale;
  }
}

// in-place row softmax over 128 columns; one wave per row
__global__ void k_softmax128(float* __restrict__ attn) {
  int row = blockIdx.x;
  float* p = attn + (long)row * 128;
  int lane = threadIdx.x & 31;
  float x0 = p[lane], x1 = p[lane + 32], x2 = p[lane + 64], x3 = p[lane + 96];
  float mx = fmaxf(fmaxf(x0, x1), fmaxf(x2, x3));
  #pragma unroll
  for (int off = 16; off > 0; off >>= 1) mx = fmaxf(mx, __shfl_xor(mx, off, 32));
  float e0 = __expf(x0 - mx), e1 = __expf(x1 - mx), e2 = __expf(x2 - mx), e3 = __expf(x3 - mx);
  float sm = e0 + e1 + e2 + e3;
  #pragma unroll
  for (int off = 16; off > 0; off >>= 1) sm += __shfl_xor(sm, off, 32);
  float inv = 1.f / sm;
  p[lane] = e0 * inv; p[lane + 32] = e1 * inv; p[lane + 64] = e2 * inv; p[lane + 96] = e3 * inv;
}

// spectral attention: scores + softmax + cosine mask + /sqrt(1.1); one wave/row
__global__ void k_attn_l(const float* __restrict__ ql, const float* __restrict__ kl,
                         float* __restrict__ attnl) {
  int row = blockIdx.x;                    // 16*28 rows
  int n = row / C_, i = row % C_;
  int lane = threadIdx.x & 31;
  float val = 0.f;
  if (lane < C_) {
    const float* qi = ql + (n * C_ + i) * UA_;
    const float* kj = kl + (n * C_ + lane) * UA_;
    float acc = 0.f;
    #pragma unroll 4
    for (int u = 0; u < UA_; ++u) acc += qi[u] * kj[u];
    val = acc * 0.10206207261596576f;      // /sqrt(96)
  }
  float mv = (lane < C_) ? val : -1e30f;
  #pragma unroll
  for (int off = 16; off > 0; off >>= 1) mv = fmaxf(mv, __shfl_xor(mv, off, 32));
  float e = (lane < C_) ? __expf(val - mv) : 0.f;
  float sm = e;
  #pragma unroll
  for (int off = 16; off > 0; off >>= 1) sm += __shfl_xor(sm, off, 32);
  if (lane < C_) {
    float mask = (__cosf(PI_ * (float)(lane - i) / 27.f) + 1.f) * 0.5f;
    attnl[(n * C_ + i) * C_ + lane] = (e / sm + mask) * 0.9534625892455922f; // /sqrt(1.1)
  }
}

// ===========================================================================
// 6) down-conv (3x3,s2,p1, C->2C) + BN + relu as implicit GEMM via WMMA
//    per n: M=56 (c2), K=252, N=16384 (x,y); 2-way N blocking
// ===========================================================================
__device__ __forceinline__ float down_B(const float* vn, int k, int x, int y) {
  int ci = k / 9;
  int rr = k % 9;
  int dx = rr / 3, dy = rr % 3;
  int ih = 2 * x - 1 + dx;
  int iw = 2 * y - 1 + dy;
  return (ih >= 0 && ih < 256 && iw >= 0 && iw < 256) ? vn[(ci * 256 + ih) * 256 + iw] : 0.f;
}

__global__ void k_down_wmma(const float* __restrict__ v, const float* __restrict__ w,
                            const float* cb, const float* sc, const float* bi,
                            const float* mn, const float* vr, float* __restrict__ pro0) {
  int wid = blockIdx.x;                 // 16 * 4 * 512
  int n = wid / 2048;
  int r = wid % 2048;
  int mt = r / 512;
  int t = r % 512;
  int x = t >> 2;
  int y0 = (t & 3) << 5;                // 32-wide column pair
  int lane = threadIdx.x & 31;
  int half = lane >> 4, l16 = lane & 15;
  int c2 = mt * 16 + l16;
  bool c2ok = c2 < C2_;
  int yc0 = y0 + l16;
  int yc1 = yc0 + 16;
  const float* vn = v + (long)n * C_ * HW_;
  v8f acc0 = {0.f, 0.f, 0.f, 0.f, 0.f, 0.f, 0.f, 0.f};
  v8f acc1 = acc0;
  for (int k0 = 0; k0 < 252; k0 += 4) {
    int ka = k0 + 2 * half;
    v2f a, b0, b1;
    a.x = c2ok ? w[c2 * 252 + ka] : 0.f;
    a.y = c2ok ? w[c2 * 252 + ka + 1] : 0.f;
    b0.x = down_B(vn, ka, x, yc0);
    b0.y = down_B(vn, ka + 1, x, yc0);
    b1.x = down_B(vn, ka, x, yc1);
    b1.y = down_B(vn, ka + 1, x, yc1);
    acc0 = wmma4(a, b0, acc0);
    acc1 = wmma4(a, b1, acc1);
  }
  #pragma unroll
  for (int j = 0; j < 8; ++j) {
    int mm = mt * 16 + j + 8 * half;
    if (mm >= C2_) continue;
    long base = (((long)n * C2_ + mm) * 128 + x) * 128;
    float y0v = fmaxf(bn_apply(acc0[j] + cb[mm], sc, bi, mn, vr, mm), 0.f);
    float y1v = fmaxf(bn_apply(acc1[j] + cb[mm], sc, bi, mn, vr, mm), 0.f);
    pro0[base + y0 + l16] = y0v;
    pro0[base + y0 + l16 + 16] = y1v;
  }
}

// ===========================================================================
// 7) out1[n,c,x,z] = sum_y pro0[n,c,x,y] * attn_y[n,z,y]   (WMMA, 2-way N)
// ===========================================================================
__global__ void k_apply_y_wmma(const float* __restrict__ pro0, const float* __restrict__ ay,
                               float* __restrict__ out1) {
  int wid = blockIdx.x;                 // 16 * 448 * 4
  int n = wid / (448 * 4);
  int r = wid % (448 * 4);
  int mt = r / 4, np = r % 4;
  int lane = threadIdx.x & 31;
  int half = lane >> 4, l16 = lane & 15;
  const float* A = pro0 + (long)n * 7168 * 128;
  const float* ayn = ay + n * 16384;
  int arow = mt * 16 + l16;
  int bcol0 = np * 32 + l16;
  int bcol1 = bcol0 + 16;
  v8f acc0 = {0.f, 0.f, 0.f, 0.f, 0.f, 0.f, 0.f, 0.f};
  v8f acc1 = acc0;
  for (int k0 = 0; k0 < 128; k0 += 4) {
    int ka = k0 + 2 * half;
    v2f a, b0, b1;
    a.x = A[arow * 128 + ka];
    a.y = A[arow * 128 + ka + 1];
    b0.x = ayn[bcol0 * 128 + ka];       // B(k=y, z) = attn_y[n, z, y]
    b0.y = ayn[bcol0 * 128 + ka + 1];
    b1.x = ayn[bcol1 * 128 + ka];
    b1.y = ayn[bcol1 * 128 + ka + 1];
    acc0 = wmma4(a, b0, acc0);
    acc1 = wmma4(a, b1, acc1);
  }
  float* O = out1 + (long)n * 7168 * 128;
  #pragma unroll
  for (int j = 0; j < 8; ++j) {
    int mm = mt * 16 + j + 8 * half;
    O[mm * 128 + np * 32 + l16] = acc0[j];
    O[mm * 128 + np * 32 + l16 + 16] = acc1[j];
  }
}

// out2[n,c,w,y] = sum_x attn_x[n,w,x] * out1[n,c,x,y]   (WMMA, 2-way N)
__global__ void k_apply_x_wmma(const float* __restrict__ ax, const float* __restrict__ out1,
                               float* __restrict__ out2) {
  int wid = blockIdx.x;                 // (16*56) * 32
  int nc = wid / 32;
  int t = wid % 32;
  int mt = t >> 2, np = t & 3;
  int lane = threadIdx.x & 31;
  int half = lane >> 4, l16 = lane & 15;
  const float* A = ax + (long)(nc / C2_) * 16384;
  const float* Bm = out1 + (long)nc * 16384;
  int arow = mt * 16 + l16;
  int bcol0 = np * 32 + l16;
  int bcol1 = bcol0 + 16;
  v8f acc0 = {0.f, 0.f, 0.f, 0.f, 0.f, 0.f, 0.f, 0.f};
  v8f acc1 = acc0;
  for (int k0 = 0; k0 < 128; k0 += 4) {
    int ka = k0 + 2 * half;
    v2f a, b0, b1;
    a.x = A[arow * 128 + ka];
    a.y = A[arow * 128 + ka + 1];
    b0.x = Bm[ka * 128 + bcol0];
    b0.y = Bm[(ka + 1) * 128 + bcol0];
    b1.x = Bm[ka * 128 + bcol1];
    b1.y = Bm[(ka + 1) * 128 + bcol1];
    acc0 = wmma4(a, b0, acc0);
    acc1 = wmma4(a, b1, acc1);
  }
  float* O = out2 + (long)nc * 16384;
  #pragma unroll
  for (int j = 0; j < 8; ++j) {
    int mm = mt * 16 + j + 8 * half;
    O[mm * 128 + np * 32 + l16] = acc0[j];
    O[mm * 128 + np * 32 + l16 + 16] = acc1[j];
  }
}

// ===========================================================================
// 8) up: ConvTranspose2d(2C->C, k=s=2): pro3[16,28,256,256]
// ===========================================================================
__global__ void k_up(const float* __restrict__ out2, const float* __restrict__ uw,
                     const float* __restrict__ ub, float* __restrict__ pro3) {
  int idx = blockIdx.x * blockDim.x + threadIdx.x;
  if (idx >= NB_ * C_ * HW_) return;
  int p = idx % HW_;
  int d = (idx / HW_) % C_;
  int n = idx / (HW_ * C_);
  int ph = p >> 8, pw = p & 255;
  int x = ph >> 1, a = ph & 1;
  int y = pw >> 1, bb = pw & 1;
  float acc = ub[d];
  #pragma unroll 4
  for (int c = 0; c < C2_; ++c)
    acc += out2[(((long)n * C2_ + c) * 128 + x) * 128 + y] * uw[(c * C_ + d) * 4 + a * 2 + bb];
  pro3[idx] = acc;
}

// 9) spectral channel mix: pro4[n,d,s] = sum_c attn_l[n,d,c] * pro3[n,c,s]
__global__ void k_specmix(const float* __restrict__ attnl, const float* __restrict__ pro3,
                          float* __restrict__ pro4) {
  int idx = blockIdx.x * blockDim.x + threadIdx.x;
  if (idx >= NB_ * C_ * HW_) return;
  int s = idx % HW_;
  int d = (idx / HW_) % C_;
  int n = idx / (HW_ * C_);
  const float* al = attnl + (n * C_ + d) * C_;
  const float* pp = pro3 + (long)n * C_ * HW_ + s;
  float acc = 0.f;
  #pragma unroll 4
  for (int c = 0; c < C_; ++c) acc += al[c] * pp[(long)c * HW_];
  pro4[idx] = acc;
}

// ===========================================================================
// 10) agg Linear(NH*C -> C) + gamma*x + residual  (WMMA, K=224, 2-way N)
// ===========================================================================
__global__ void k_agg_wmma(const float* __restrict__ pro4, const float* __restrict__ aw,
                           const float* __restrict__ ab, const float* __restrict__ gamma,
                           const float* __restrict__ feat, float* __restrict__ outp) {
  int wid = blockIdx.x;                 // B * 2 * 2048
  int b = wid / 4096;
  int r = wid % 4096;
  int mt = r / 2048;
  int np = r % 2048;
  int lane = threadIdx.x & 31;
  int half = lane >> 4, l16 = lane & 15;
  int arow = mt * 16 + l16;
  bool arok = arow < C_;
  int bcol0 = np * 32 + l16;            // hw columns
  int bcol1 = bcol0 + 16;
  v8f acc0 = {0.f, 0.f, 0.f, 0.f, 0.f, 0.f, 0.f, 0.f};
  v8f acc1 = acc0;
  for (int k0 = 0; k0 < 224; k0 += 4) {
    int ka = k0 + 2 * half;
    v2f a, b0, b1;
    a.x = arok ? aw[arow * 224 + ka] : 0.f;
    a.y = arok ? aw[arow * 224 + ka + 1] : 0.f;
    int h0 = ka / C_, c0 = ka % C_;
    int h1 = (ka + 1) / C_, c1 = (ka + 1) % C_;
    long r0 = (((long)(h0 * B_ + b) * C_) + c0) * HW_;
    long r1 = (((long)(h1 * B_ + b) * C_) + c1) * HW_;
    b0.x = pro4[r0 + bcol0];
    b0.y = pro4[r1 + bcol0];
    b1.x = pro4[r0 + bcol1];
    b1.y = pro4[r1 + bcol1];
    acc0 = wmma4(a, b0, acc0);
    acc1 = wmma4(a, b1, acc1);
  }
  float g = gamma[0];
  #pragma unroll
  for (int j = 0; j < 8; ++j) {
    int mm = mt * 16 + j + 8 * half;
    if (mm >= C_) continue;
    long o0 = ((long)b * C_ + mm) * HW_ + bcol0;
    outp[o0] = g * (acc0[j] + ab[mm]) + feat[o0];
    outp[o0 + 16] = g * (acc1[j] + ab[mm]) + feat[o0 + 16];
  }
}

// ===========================================================================
// launch
// ===========================================================================
extern "C" void kernel_launch(void* const* d_in, const int* in_sizes, int n_in,
                              void* d_out, int out_size, void* d_ws, size_t ws_size,
                              hipStream_t stream) {
  auto in = [&](int i) { return (const float*)d_in[i]; };

  // ---- input indices (recursive insertion-order flatten of setup_inputs) ----
  const int I_IMG = 0, I_PW = 1, I_PBN = 2;                 // pbn: s,b,m,v -> 2..5
  const int I_QX = 6, I_QY = 20, I_KX = 34, I_KY = 48;      // spat blocks, 14 each
  const int I_QL = 62, I_KL = 76;                           // spec blocks, 14 each
  const int I_VW = 90, I_VBN = 91;                          // vbn: 91..94
  const int I_DCW = 95, I_DCB = 96, I_DBN = 97;             // dbn: 97..100
  const int I_UPW = 101, I_UPB = 102, I_AW = 103, I_AB = 104, I_G = 105;
  (void)in_sizes; (void)n_in; (void)out_size; (void)ws_size;

  // ---- workspace layout (floats) ----
  float* ws = (float*)d_ws;
  float* feat = ws;                       // 3,670,016
  float* qx = feat + 3670016;             // 196,608 x4
  float* qy = qx + 196608;
  float* kx = qy + 196608;
  float* ky = kx + 196608;
  float* ql = ky + 196608;                // 43,008 x2
  float* kl = ql + 43008;
  float* s1 = kl + 43008;                 // 524,288
  float* s2 = s1 + 524288;                // 131,072
  float* t1 = s2 + 131072;                // 889,056
  float* t2 = t1 + 889056;                // 208,376
  float* bufA = t2 + 208376;              // 29,360,128 : v  -> pro3
  float* bufB = bufA + 29360128;          // 14,680,064 : pro0 -> out2
  float* bufC = bufB + 14680064;          // 29,360,128 : out1 -> pro4

  float* out = (float*)d_out;             // [2,28,256,256]
  float* attnx = out + 3670016;           // [16,128,128]
  float* attny = attnx + 262144;
  float* attnl = attny + 262144;          // [16,28,28]

  // 1) preproc
  k_preproc<<<(B_ * C_ * HW_ + 255) / 256, 256, 0, stream>>>(
      in(I_IMG), in(I_PW), in(I_PBN), in(I_PBN + 1), in(I_PBN + 2), in(I_PBN + 3), feat);

  // 2) spatial branches (qx, qy, kx, ky)
  struct SpatJob { int base; float* dst; int tr; };
  SpatJob sj[4] = {{I_QX, qx, 0}, {I_QY, qy, 1}, {I_KX, kx, 0}, {I_KY, ky, 1}};
  for (int i = 0; i < 4; ++i) {
    int p = sj[i].base;
    k_sp_conv1<<<(B_ * NH_ * 256 * 128 + 255) / 256, 256, 0, stream>>>(
        feat, in(p + 0), in(p + 1), in(p + 2), in(p + 3), in(p + 4), in(p + 5), s1, sj[i].tr);
    k_sp_conv2<<<(B_ * NH_ * 128 * 64 + 255) / 256, 256, 0, stream>>>(
        s1, in(p + 6), in(p + 7), in(p + 8), in(p + 9), in(p + 10), in(p + 11), s2);
    k_sp_fc_wmma<<<128 * 3, 32, 0, stream>>>(s2, in(p + 12), in(p + 13), sj[i].dst);
  }

  // 3) spectral branches (qlam, klam)
  struct SpecJob { int base; float* dst; };
  SpecJob cj[2] = {{I_QL, ql}, {I_KL, kl}};
  for (int i = 0; i < 2; ++i) {
    int p = cj[i].base;
    k_spec_conv<<<(B_ * C_ * 126 * 126 + 255) / 256, 256, 0, stream>>>(
        feat, 256, 126, in(p + 0), in(p + 1), in(p + 2), in(p + 3), in(p + 4), in(p + 5), t1);
    k_spec_conv<<<(B_ * C_ * 61 * 61 + 255) / 256, 256, 0, stream>>>(
        t1, 126, 61, in(p + 6), in(p + 7), in(p + 8), in(p + 9), in(p + 10), in(p + 11), t2);
    k_spec_fc_wmma<<<4 * 24, 32, 0, stream>>>(t2, in(p + 12), in(p + 13), cj[i].dst);
  }

  // 4) value conv (head-major v) into bufA
  k_value<<<(NB_ * C_ * HW_ + 255) / 256, 256, 0, stream>>>(
      feat, in(I_VW), in(I_VBN), in(I_VBN + 1), in(I_VBN + 2), in(I_VBN + 3), bufA);

  // 5) attention maps (written directly to d_out regions)
  k_scores_wmma<<<16 * 32, 32, 0, stream>>>(qx, kx, attnx);
  k_softmax128<<<16 * 128, 32, 0, stream>>>(attnx);
  k_scores_wmma<<<16 * 32, 32, 0, stream>>>(qy, ky, attny);
  k_softmax128<<<16 * 128, 32, 0, stream>>>(attny);
  k_attn_l<<<NB_ * C_, 32, 0, stream>>>(ql, kl, attnl);

  // 6) down-conv implicit GEMM: bufA (v) -> bufB (pro0)
  k_down_wmma<<<16 * 4 * 512, 32, 0, stream>>>(
      bufA, in(I_DCW), in(I_DCB), in(I_DBN), in(I_DBN + 1), in(I_DBN + 2), in(I_DBN + 3), bufB);

  // 7) attention application: pro0 x A_y^T -> bufC (out1); A_x x out1 -> bufB (out2)
  k_apply_y_wmma<<<16 * 448 * 4, 32, 0, stream>>>(bufB, attny, bufC);
  k_apply_x_wmma<<<16 * C2_ * 32, 32, 0, stream>>>(attnx, bufC, bufB);

  // 8) transposed-conv up-sample: bufB (out2) -> bufA (pro3)
  k_up<<<(NB_ * C_ * HW_ + 255) / 256, 256, 0, stream>>>(bufB, in(I_UPW), in(I_UPB), bufA);

  // 9) spectral channel mix: attn_l x pro3 -> bufC (pro4)
  k_specmix<<<(NB_ * C_ * HW_ + 255) / 256, 256, 0, stream>>>(attnl, bufA, bufC);

  // 10) aggregation + gamma*x + residual -> out
  k_agg_wmma<<<B_ * 2 * 2048, 32, 0, stream>>>(
      bufC, in(I_AW), in(I_AB), in(I_G), feat, out);
}